// CustomStellarEncoder_1_7000796693091
// MI455X (gfx1250) — compile-verified
//
#include <hip/hip_runtime.h>
#include <hip/hip_bf16.h>

typedef __attribute__((ext_vector_type(2))) float v2f;
typedef __attribute__((ext_vector_type(8))) float v8f;

#define HIDC 128
#define KPANEL 128   // K-panel staged in LDS: 128 x 128 f32 = 64 KB

// ---- CDNA5 async global->LDS copies (ASYNCcnt path), inline asm (portable) --
__device__ __forceinline__ void async_ld_b32(unsigned lds_off, const float* gptr) {
    asm volatile("global_load_async_to_lds_b32 %0, %1, off"
                 :: "v"(lds_off), "v"(gptr)
                 : "memory");
}
__device__ __forceinline__ void wait_asynccnt0() {
    asm volatile("s_wait_asynccnt 0" ::: "memory");
}

// ---------------------------------------------------------------------------
// GEMM: C[N,128] (=|+=) A[N,K] @ W[K,128] (+ bias), optional per-row scale on A.
// fp32 WMMA 16x16x4. One wave -> 16 rows x 128 cols (8 tiles). 8 waves/block.
// W is staged per-128-K panel into LDS with async B32 loads using a K-pair-
// interleaved layout: element (k,c) -> dword (k>>1)*256 + 2c + (k&1), so each
// lane's WMMA B fragment (W[k,c],W[k+1,c]) is one aligned ds_load_b64.
// A rows stream from global. K must be a multiple of 128.
// ---------------------------------------------------------------------------
__global__ __launch_bounds__(256) void sage_gemm_wmma(
    const float* __restrict__ A, const float* __restrict__ rowscale,
    const float* __restrict__ W, const float* __restrict__ bias,
    float* __restrict__ C, int Nrows, int K, int accFlag)
{
    extern __shared__ float lw[];                  // KPANEL/2 pairs x 128 x v2f (64 KB)

    const int lane = threadIdx.x & 31;
    const int wave = threadIdx.x >> 5;
    const int m0 = (blockIdx.x * 8 + wave) * 16;   // may exceed Nrows; fully masked

    const int half = lane >> 4;                    // 0: K={0,1}, 1: K={2,3}
    const int l15  = lane & 15;

    const int arow  = m0 + l15;
    const int arowc = (arow < Nrows) ? arow : (Nrows - 1);
    float ascale = (arow < Nrows) ? 1.0f : 0.0f;
    if (rowscale) ascale *= rowscale[arowc];

    v8f acc[8];
    if (accFlag) {
        #pragma unroll
        for (int t = 0; t < 8; ++t) {
            const int col = t * 16 + l15;
            #pragma unroll
            for (int v = 0; v < 8; ++v) {
                const int row = m0 + v + half * 8;
                acc[t][v] = (row < Nrows) ? C[(size_t)row * HIDC + col] : 0.0f;
            }
        }
    } else {
        #pragma unroll
        for (int t = 0; t < 8; ++t) {
            const float bv = bias ? bias[t * 16 + l15] : 0.0f;
            #pragma unroll
            for (int v = 0; v < 8; ++v) acc[t][v] = bv;
        }
    }

    const float* __restrict__ Arow = A + (size_t)arowc * K;
    const unsigned ldsbase = (unsigned)(uintptr_t)lw;

    for (int kb = 0; kb < K; kb += KPANEL) {
        if (kb) __syncthreads();                   // LDS panel reuse fence
        // Async-stage W[kb : kb+128) x 128 into LDS, pair-interleaved.
        {
            const float* wsrc = W + (size_t)kb * HIDC;
            #pragma unroll 8
            for (int i = threadIdx.x; i < KPANEL * HIDC; i += 256) {
                const unsigned k = (unsigned)i >> 7;
                const unsigned c = (unsigned)i & (HIDC - 1);
                const unsigned dst =
                    ldsbase + ((((k >> 1) * (HIDC * 2)) + c * 2u + (k & 1u)) << 2);
                async_ld_b32(dst, wsrc + i);
            }
        }
        if (kb + KPANEL < K)                       // prefetch next A panel
            __builtin_prefetch(Arow + kb + KPANEL, 0, 0);
        wait_asynccnt0();
        __syncthreads();

        #pragma unroll 2
        for (int kk = 0; kk < KPANEL; kk += 4) {
            const v2f av = *(const v2f*)(Arow + kb + kk + half * 2);
            v2f a;
            a.x = av.x * ascale;
            a.y = av.y * ascale;
            const int p = (kk >> 1) + half;        // K-pair index for this half-wave
            const float* bbase = lw + (size_t)p * (HIDC * 2) + l15 * 2;
            v2f bf[8];
            #pragma unroll
            for (int t = 0; t < 8; ++t)            // 8x ds_load_b64, issued up front
                bf[t] = *(const v2f*)(bbase + t * 32);
            #pragma unroll
            for (int t = 0; t < 8; ++t)
                acc[t] = __builtin_amdgcn_wmma_f32_16x16x4_f32(
                    false, a, false, bf[t], (short)0, acc[t], false, false);
        }
    }

    #pragma unroll
    for (int t = 0; t < 8; ++t) {
        const int col = t * 16 + l15;
        #pragma unroll
        for (int v = 0; v < 8; ++v) {
            const int row = m0 + v + half * 8;
            if (row < Nrows) C[(size_t)row * HIDC + col] = acc[t][v];
        }
    }
}

// ---------------------------------------------------------------------------
// BatchNorm statistics: per-column sum and sum of squares over N rows.
// ---------------------------------------------------------------------------
__global__ __launch_bounds__(256) void sage_bn_stats(
    const float* __restrict__ z, float* __restrict__ sums,
    float* __restrict__ sumsq, int Nrows)
{
    const int col = threadIdx.x & (HIDC - 1);
    const int sub = threadIdx.x >> 7;
    float s = 0.0f, q = 0.0f;
    for (int r = blockIdx.x * 2 + sub; r < Nrows; r += gridDim.x * 2) {
        const float v = z[(size_t)r * HIDC + col];
        s += v;
        q += v * v;
    }
    __shared__ float ls[256];
    __shared__ float lq[256];
    ls[threadIdx.x] = s; lq[threadIdx.x] = q;
    __syncthreads();
    if (threadIdx.x < HIDC) {
        unsafeAtomicAdd(&sums[col],  ls[threadIdx.x] + ls[threadIdx.x + HIDC]);
        unsafeAtomicAdd(&sumsq[col], lq[threadIdx.x] + lq[threadIdx.x + HIDC]);
    }
}

// ab[0:128] = gamma*rsqrt(var+eps); ab[128:256] = beta - mean*that
__global__ void sage_bn_final(const float* __restrict__ sums,
                              const float* __restrict__ sumsq,
                              const float* __restrict__ g,
                              const float* __restrict__ be,
                              float* __restrict__ ab, float invN)
{
    const int c = threadIdx.x;
    if (c >= HIDC) return;
    const float mean = sums[c] * invN;
    const float var  = sumsq[c] * invN - mean * mean;
    const float a = g[c] * rsqrtf(var + 1e-5f);
    ab[c] = a;
    ab[HIDC + c] = be[c] - mean * a;
}

__global__ __launch_bounds__(256) void sage_bn_apply(
    const float* __restrict__ z, const float* __restrict__ ab,
    float* __restrict__ y, long long total, int relu)
{
    const long long i = (long long)blockIdx.x * blockDim.x + threadIdx.x;
    if (i >= total) return;
    const int c = (int)(i & (HIDC - 1));
    float v = z[i] * ab[c] + ab[HIDC + c];
    if (relu) v = fmaxf(v, 0.0f);
    y[i] = v;
}

// ---------------------------------------------------------------------------
// Edge scatter: one wave32 per edge; each lane moves 4 floats (float4 gather,
// 4x global_atomic_add_f32 scatter). Gather source and scatter target are both
// 25.6 MB -> resident in the 192 MB L2, so this runs at L2 (not HBM) speed.
// ---------------------------------------------------------------------------
__global__ __launch_bounds__(256) void sage_scatter(
    const float* __restrict__ feat, const int* __restrict__ ei,
    float* __restrict__ agg, float* __restrict__ deg, int Eedges)
{
    const long long gid = (long long)blockIdx.x * blockDim.x + threadIdx.x;
    const int e = (int)(gid >> 5);
    const int lane = (int)(gid & 31);
    if (e >= Eedges) return;
    const int src = ei[e];
    const int dst = ei[Eedges + e];
    if (lane == 0) unsafeAtomicAdd(&deg[dst], 1.0f);
    const float4 v = ((const float4*)(feat + (size_t)src * HIDC))[lane];
    float* base = agg + (size_t)dst * HIDC + lane * 4;
    unsafeAtomicAdd(base + 0, v.x);
    unsafeAtomicAdd(base + 1, v.y);
    unsafeAtomicAdd(base + 2, v.z);
    unsafeAtomicAdd(base + 3, v.w);
}

// deg[i] := 1 / max(deg[i], 1)
__global__ void sage_deg_rinv(float* __restrict__ deg, int n)
{
    const int i = blockIdx.x * blockDim.x + threadIdx.x;
    if (i < n) deg[i] = 1.0f / fmaxf(deg[i], 1.0f);
}

// ---------------------------------------------------------------------------
extern "C" void kernel_launch(void* const* d_in, const int* in_sizes, int n_in,
                              void* d_out, int out_size, void* d_ws, size_t ws_size,
                              hipStream_t stream)
{
    const float* x     = (const float*)d_in[0];
    const int*   ei    = (const int*)  d_in[1];
    const float* W_in  = (const float*)d_in[2];
    const float* b_in  = (const float*)d_in[3];
    const float* g1    = (const float*)d_in[4];
    const float* be1   = (const float*)d_in[5];
    const float* W_hid = (const float*)d_in[6];
    const float* b_hid = (const float*)d_in[7];
    const float* g2    = (const float*)d_in[8];
    const float* be2   = (const float*)d_in[9];
    const float* Wl1   = (const float*)d_in[10];
    const float* bl1   = (const float*)d_in[11];
    const float* Wr1   = (const float*)d_in[12];
    const float* g3    = (const float*)d_in[13];
    const float* be3   = (const float*)d_in[14];
    const float* Wl2   = (const float*)d_in[15];
    const float* bl2   = (const float*)d_in[16];
    const float* Wr2   = (const float*)d_in[17];
    const float* g4    = (const float*)d_in[18];
    const float* be4   = (const float*)d_in[19];

    const int K1 = in_sizes[2] / HIDC;          // 256
    const int N  = in_sizes[0] / K1;            // 50000
    const int E  = in_sizes[1] / 2;             // 1600000
    const long long NF = (long long)N * HIDC;

    float* out0 = (float*)d_out;                // feat
    float* out1 = out0 + NF;                    // out

    float* bufA  = (float*)d_ws;                // N*128 (z1 / agg)
    float* bufB  = bufA + NF;                   // N*128 (feat1 / z3 / out3)
    float* deg   = bufB + NF;                   // N
    float* sums  = deg + N;                     // 128
    float* sumsq = sums + HIDC;                 // 128
    float* ab    = sumsq + HIDC;                // 256

    const int gemmBlocks  = (N + 127) / 128;
    const int applyBlocks = (int)((NF + 255) / 256);
    const int scatBlocks  = (E + 7) / 8;        // 8 edges (waves) per 256-thr block
    const size_t ldsBytes = (size_t)KPANEL * HIDC * sizeof(float);  // 64 KB
    const float invN = 1.0f / (float)N;

    auto bn = [&](const float* z, float* y, const float* g, const float* be, int relu) {
        (void)hipMemsetAsync(sums, 0, 2 * HIDC * sizeof(float), stream);
        sage_bn_stats<<<512, 256, 0, stream>>>(z, sums, sumsq, N);
        sage_bn_final<<<1, 128, 0, stream>>>(sums, sumsq, g, be, ab, invN);
        sage_bn_apply<<<applyBlocks, 256, 0, stream>>>(z, ab, y, NF, relu);
    };

    // Stage 1: z1 = x @ W_in + b_in ; feat1 = relu(bn(z1))
    sage_gemm_wmma<<<gemmBlocks, 256, ldsBytes, stream>>>(x, nullptr, W_in, b_in, bufA, N, K1, 0);
    bn(bufA, bufB, g1, be1, 1);

    // Stage 2: z2 = feat1 @ W_hid + b_hid ; feat = relu(bn(z2))  -> out0
    sage_gemm_wmma<<<gemmBlocks, 256, ldsBytes, stream>>>(bufB, nullptr, W_hid, b_hid, out0, N, HIDC, 0);
    bn(out0, out0, g2, be2, 1);

    // Stage 3: SAGE1 on feat(out0) -> out3 (bufB)
    (void)hipMemsetAsync(bufA, 0, (size_t)NF * sizeof(float), stream);
    (void)hipMemsetAsync(deg, 0, (size_t)N * sizeof(float), stream);
    sage_scatter<<<scatBlocks, 256, 0, stream>>>(out0, ei, bufA, deg, E);
    sage_deg_rinv<<<(N + 255) / 256, 256, 0, stream>>>(deg, N);
    sage_gemm_wmma<<<gemmBlocks, 256, ldsBytes, stream>>>(bufA, deg, Wl1, bl1, bufB, N, HIDC, 0);
    sage_gemm_wmma<<<gemmBlocks, 256, ldsBytes, stream>>>(out0, nullptr, Wr1, nullptr, bufB, N, HIDC, 1);
    bn(bufB, bufB, g3, be3, 0);

    // Stage 4: SAGE2 on out3(bufB) -> out1
    (void)hipMemsetAsync(bufA, 0, (size_t)NF * sizeof(float), stream);
    (void)hipMemsetAsync(deg, 0, (size_t)N * sizeof(float), stream);
    sage_scatter<<<scatBlocks, 256, 0, stream>>>(bufB, ei, bufA, deg, E);
    sage_deg_rinv<<<(N + 255) / 256, 256, 0, stream>>>(deg, N);
    sage_gemm_wmma<<<gemmBlocks, 256, ldsBytes, stream>>>(bufA, deg, Wl2, bl2, out1, N, HIDC, 0);
    sage_gemm_wmma<<<gemmBlocks, 256, ldsBytes, stream>>>(bufB, nullptr, Wr2, nullptr, out1, N, HIDC, 1);
    bn(out1, out1, g4, be4, 0);

    (void)n_in; (void)out_size; (void)ws_size;
}